// GCN_76020921139207
// MI455X (gfx1250) — compile-verified
//
#include <hip/hip_runtime.h>

typedef float v2f __attribute__((ext_vector_type(2)));
typedef float v8f __attribute__((ext_vector_type(8)));

// ---------------------------------------------------------------------------
// Utility: zero-fill a float buffer (graph-capture-safe, no hipMemsetAsync).
// ---------------------------------------------------------------------------
__global__ void gcn_zero_f32(float* __restrict__ p, long long n) {
    long long i = (long long)blockIdx.x * blockDim.x + threadIdx.x;
    if (i < n) p[i] = 0.0f;
}

// ---------------------------------------------------------------------------
// Degree: deg[dst] += 1 per edge (float atomics; deg buffer pre-zeroed).
// ---------------------------------------------------------------------------
__global__ void gcn_degree(const long long* __restrict__ dst,
                           float* __restrict__ deg, int E) {
    int e = blockIdx.x * blockDim.x + threadIdx.x;
    if (e < E) atomicAdd(&deg[(int)dst[e]], 1.0f);
}

// dinv[i] = rsqrt(deg[i] + 1)  (self-loop), in place over deg.
__global__ void gcn_dinv(float* __restrict__ deg, int N) {
    int i = blockIdx.x * blockDim.x + threadIdx.x;
    if (i < N) deg[i] = __frsqrt_rn(deg[i] + 1.0f);
}

// ---------------------------------------------------------------------------
// GEMM1: XW = X (Mx128) * W1 (128x128), f32 WMMA 16x16x4.
// One wave per 16x16 output tile; block = 8 waves covers full 128-col span.
// A layout (16x4 f32): lanes 0-15 -> M=lane, VGPRs {K0,K1}; lanes 16-31 -> {K2,K3}.
// B layout (4x16 f32): lanes 0-15 -> N=lane, VGPRs {K0,K1}; lanes 16-31 -> {K2,K3}.
// C/D layout: lane<16 -> M=v, N=lane; lane>=16 -> M=8+v, N=lane-16.
// ---------------------------------------------------------------------------
__global__ __launch_bounds__(256) void gcn_gemm1_wmma(
        const float* __restrict__ X, const float* __restrict__ W,
        float* __restrict__ XW, int M) {
    const int lane  = threadIdx.x & 31;
    const int wave  = threadIdx.x >> 5;          // N-tile 0..7
    const int row0  = blockIdx.x * 16;
    const int col0  = wave * 16;
    const int l15   = lane & 15;
    const int koff  = (lane >> 4) * 2;           // 0 or 2
    int arow = row0 + l15;
    if (arow >= M) arow = M - 1;                 // clamped load; store guarded
    const int bcol = col0 + l15;

    v8f c = {};
#pragma unroll
    for (int k = 0; k < 128; k += 4) {
        v2f a = *(const v2f*)(X + (size_t)arow * 128 + k + koff);
        v2f b;
        b[0] = W[(size_t)(k + koff    ) * 128 + bcol];
        b[1] = W[(size_t)(k + koff + 1) * 128 + bcol];
        c = __builtin_amdgcn_wmma_f32_16x16x4_f32(
                /*neg_a=*/false, a, /*neg_b=*/false, b,
                /*c_mod=*/(short)0, c, /*reuse_a=*/false, /*reuse_b=*/false);
    }

    const int crow0 = row0 + ((lane >> 4) << 3); // +0 or +8
    const int ccol  = col0 + l15;
#pragma unroll
    for (int v = 0; v < 8; ++v) {
        int r = crow0 + v;
        if (r < M) XW[(size_t)r * 128 + ccol] = c[v];
    }
}

// ---------------------------------------------------------------------------
// Edge aggregation, 128 channels: one wave per edge, lane handles 4 channels.
// agg[dst] += XW[src] * dinv[src]*dinv[dst].  XW/agg live in L2 (51 MB each).
// ---------------------------------------------------------------------------
__global__ __launch_bounds__(256) void gcn_agg128(
        const long long* __restrict__ src, const long long* __restrict__ dst,
        const float* __restrict__ dinv, const float* __restrict__ xw,
        float* __restrict__ agg, int E) {
    int wid = (int)(((long long)blockIdx.x * blockDim.x + threadIdx.x) >> 5);
    if (wid >= E) return;
    const int lane = threadIdx.x & 31;
    const int s = (int)src[wid];
    const int d = (int)dst[wid];
    const float norm = dinv[s] * dinv[d];
    const float4 v = *(const float4*)(xw + (size_t)s * 128 + lane * 4);
    float* a = agg + (size_t)d * 128 + lane * 4;
    atomicAdd(a + 0, v.x * norm);
    atomicAdd(a + 1, v.y * norm);
    atomicAdd(a + 2, v.z * norm);
    atomicAdd(a + 3, v.w * norm);
}

// ---------------------------------------------------------------------------
// Epilogue 1: h = relu(agg1 + XW*dinv^2 + b1), written in place over agg1.
// Thread handles 4 channels.
// ---------------------------------------------------------------------------
__global__ void gcn_post1(float* __restrict__ agg, const float* __restrict__ xw,
                          const float* __restrict__ dinv,
                          const float* __restrict__ b1, int N) {
    long long idx = (long long)blockIdx.x * blockDim.x + threadIdx.x;
    int node = (int)(idx >> 5);
    if (node >= N) return;
    int c4 = (int)(idx & 31) * 4;
    float sii = dinv[node] * dinv[node];
    float4 a = *(float4*)(agg + (size_t)node * 128 + c4);
    float4 w = *(const float4*)(xw + (size_t)node * 128 + c4);
    float4 b = *(const float4*)(b1 + c4);
    float4 h;
    h.x = fmaxf(fmaf(w.x, sii, a.x) + b.x, 0.0f);
    h.y = fmaxf(fmaf(w.y, sii, a.y) + b.y, 0.0f);
    h.z = fmaxf(fmaf(w.z, sii, a.z) + b.z, 0.0f);
    h.w = fmaxf(fmaf(w.w, sii, a.w) + b.w, 0.0f);
    *(float4*)(agg + (size_t)node * 128 + c4) = h;
}

// ---------------------------------------------------------------------------
// GEMM2: p = h (Nx128) * W2 (128x2). Thread per node; W2 stays in cache.
// ---------------------------------------------------------------------------
__global__ void gcn_gemm2(const float* __restrict__ h,
                          const float* __restrict__ W2,
                          float* __restrict__ p, int N) {
    int i = blockIdx.x * blockDim.x + threadIdx.x;
    if (i >= N) return;
    float s0 = 0.0f, s1 = 0.0f;
#pragma unroll 4
    for (int k = 0; k < 128; k += 4) {
        float4 hv = *(const float4*)(h + (size_t)i * 128 + k);
        s0 = fmaf(hv.x, W2[(k + 0) * 2 + 0], s0);
        s1 = fmaf(hv.x, W2[(k + 0) * 2 + 1], s1);
        s0 = fmaf(hv.y, W2[(k + 1) * 2 + 0], s0);
        s1 = fmaf(hv.y, W2[(k + 1) * 2 + 1], s1);
        s0 = fmaf(hv.z, W2[(k + 2) * 2 + 0], s0);
        s1 = fmaf(hv.z, W2[(k + 2) * 2 + 1], s1);
        s0 = fmaf(hv.w, W2[(k + 3) * 2 + 0], s0);
        s1 = fmaf(hv.w, W2[(k + 3) * 2 + 1], s1);
    }
    p[(size_t)i * 2 + 0] = s0;
    p[(size_t)i * 2 + 1] = s1;
}

// ---------------------------------------------------------------------------
// Edge aggregation, 2 channels: thread per edge (multiply-first trick makes
// layer-2 scatter 64x cheaper than aggregating 128 channels).
// ---------------------------------------------------------------------------
__global__ void gcn_agg2(const long long* __restrict__ src,
                         const long long* __restrict__ dst,
                         const float* __restrict__ dinv,
                         const float* __restrict__ p,
                         float* __restrict__ agg, int E) {
    int e = blockIdx.x * blockDim.x + threadIdx.x;
    if (e >= E) return;
    const int s = (int)src[e];
    const int d = (int)dst[e];
    const float norm = dinv[s] * dinv[d];
    atomicAdd(agg + (size_t)d * 2 + 0, p[(size_t)s * 2 + 0] * norm);
    atomicAdd(agg + (size_t)d * 2 + 1, p[(size_t)s * 2 + 1] * norm);
}

// Final: out = agg2 + p*dinv^2 + b2 (fully rewrites d_out).
__global__ void gcn_final(const float* __restrict__ agg,
                          const float* __restrict__ p,
                          const float* __restrict__ dinv,
                          const float* __restrict__ b2,
                          float* __restrict__ out, int N) {
    int i = blockIdx.x * blockDim.x + threadIdx.x;
    if (i >= N) return;
    float sii = dinv[i] * dinv[i];
    out[(size_t)i * 2 + 0] = fmaf(p[(size_t)i * 2 + 0], sii, agg[(size_t)i * 2 + 0]) + b2[0];
    out[(size_t)i * 2 + 1] = fmaf(p[(size_t)i * 2 + 1], sii, agg[(size_t)i * 2 + 1]) + b2[1];
}

// ---------------------------------------------------------------------------
extern "C" void kernel_launch(void* const* d_in, const int* in_sizes, int n_in,
                              void* d_out, int out_size, void* d_ws, size_t ws_size,
                              hipStream_t stream) {
    const float*     x   = (const float*)d_in[0];
    const long long* ei  = (const long long*)d_in[1];
    const float*     W1  = (const float*)d_in[2];
    const float*     b1  = (const float*)d_in[3];
    const float*     W2  = (const float*)d_in[4];
    const float*     b2  = (const float*)d_in[5];

    const int N = in_sizes[0] / 128;   // 100000
    const int E = in_sizes[1] / 2;     // 1600000
    const long long* src = ei;
    const long long* dst = ei + E;

    // Workspace layout (floats), 16-float aligned sections.
    float* ws = (float*)d_ws;
    const size_t nPad  = ((size_t)N + 255) & ~(size_t)255;
    const size_t nfPad = (size_t)N * 128;
    float* deg  = ws;                       // N      (becomes dinv)
    float* xw   = deg  + nPad;              // N*128
    float* agg1 = xw   + nfPad;             // N*128  (becomes h)
    float* p    = agg1 + nfPad;             // N*2
    float* agg2 = p    + 2 * nPad;          // N*2

    const int B = 256;

    // 1) degree -> dinv
    gcn_zero_f32<<<(int)((N + B - 1) / B), B, 0, stream>>>(deg, N);
    gcn_degree<<<(E + B - 1) / B, B, 0, stream>>>(dst, deg, E);
    gcn_dinv<<<(N + B - 1) / B, B, 0, stream>>>(deg, N);

    // 2) XW = X*W1 via f32 WMMA
    gcn_gemm1_wmma<<<(N + 15) / 16, B, 0, stream>>>(x, W1, xw, N);

    // 3) agg1 = scatter(norm * XW[src] -> dst)
    long long nf = (long long)N * 128;
    gcn_zero_f32<<<(int)((nf + B - 1) / B), B, 0, stream>>>(agg1, nf);
    gcn_agg128<<<(E * 32 + B - 1) / B, B, 0, stream>>>(src, dst, deg, xw, agg1, E);

    // 4) h = relu(agg1 + XW*dinv^2 + b1)   (in place over agg1)
    long long nt = (long long)N * 32;
    gcn_post1<<<(int)((nt + B - 1) / B), B, 0, stream>>>(agg1, xw, deg, b1, N);

    // 5) p = h*W2 ; agg2 = scatter(norm * p[src] -> dst)
    gcn_gemm2<<<(N + B - 1) / B, B, 0, stream>>>(agg1, W2, p, N);
    gcn_zero_f32<<<(int)((2LL * N + B - 1) / B), B, 0, stream>>>(agg2, 2LL * N);
    gcn_agg2<<<(E + B - 1) / B, B, 0, stream>>>(src, dst, deg, p, agg2, E);

    // 6) out = agg2 + p*dinv^2 + b2
    gcn_final<<<(N + B - 1) / B, B, 0, stream>>>(agg2, p, deg, b2, (float*)d_out, N);
}